// LayerStateAggregator_7902739824951
// MI455X (gfx1250) — compile-verified
//
#include <hip/hip_runtime.h>
#include <math.h>

// Problem shape (fixed by reference)
#define B    32
#define S    2048
#define D    1024
#define DIN  512
#define DOUT 1024
#define SPLIT 32          // sequence splits for pass 2
#define EPSV 1e-6f

typedef __attribute__((ext_vector_type(2))) float v2f;
typedef __attribute__((ext_vector_type(8))) float v8f;

// ---------------------------------------------------------------------------
// Kernel 1: qkw[d] = ln_w[d] * sum_i Wk[d,i] * q[i]   (one wave per row d)
// ---------------------------------------------------------------------------
__global__ void k_qkw(const float* __restrict__ Wk, const float* __restrict__ q,
                      const float* __restrict__ ln_w, float* __restrict__ qkw) {
    const int wave = threadIdx.x >> 5;
    const int lane = threadIdx.x & 31;
    const int d = blockIdx.x * 8 + wave;           // grid = 128 blocks * 8 waves
    const float* row = Wk + (size_t)d * DIN;
    float s = 0.f;
    for (int i = lane; i < DIN; i += 32) s += row[i] * q[i];
    #pragma unroll
    for (int off = 16; off; off >>= 1) s += __shfl_xor(s, off, 32);
    if (lane == 0) qkw[d] = s * ln_w[d];
}

// ---------------------------------------------------------------------------
// Kernel 2: streaming pass 1 over x. One wave per (b,s) row of 1024 floats.
// Fused: sumsq (RMSNorm) + dot(x, qkw) -> r[row], score[row]
// ---------------------------------------------------------------------------
__global__ void k_pass1(const float* __restrict__ x, const float* __restrict__ qkw,
                        float* __restrict__ r, float* __restrict__ score) {
    const int wave = threadIdx.x >> 5;
    const int lane = threadIdx.x & 31;
    const size_t row = (size_t)blockIdx.x * 8 + wave;   // grid = 8192 -> 65536 rows
    const float4* xp = (const float4*)(x + row * (size_t)D);
    const float4* wp = (const float4*)qkw;
    float ss = 0.f, dt = 0.f;
    #pragma unroll
    for (int it = 0; it < D / (4 * 32); ++it) {          // 8 iterations, 128B/lane-coalesced
        float4 v = xp[it * 32 + lane];
        float4 w = wp[it * 32 + lane];
        ss += v.x * v.x + v.y * v.y + v.z * v.z + v.w * v.w;
        dt += v.x * w.x + v.y * w.y + v.z * w.z + v.w * w.w;
    }
    #pragma unroll
    for (int off = 16; off; off >>= 1) {
        ss += __shfl_xor(ss, off, 32);
        dt += __shfl_xor(dt, off, 32);
    }
    if (lane == 0) {
        float rv = rsqrtf(ss * (1.0f / D) + EPSV);
        r[row] = rv;
        score[row] = rv * dt;
    }
}

// ---------------------------------------------------------------------------
// Kernel 3: softmax over S per batch; writes c[b,s] = attn[b,s] * r[b,s]
// ---------------------------------------------------------------------------
__global__ void k_softmax(const float* __restrict__ score, const float* __restrict__ r,
                          float* __restrict__ c) {
    __shared__ float red[256];
    const int b = blockIdx.x;
    const int t = threadIdx.x;
    const float* sc = score + (size_t)b * S;

    float m = -3.4e38f;
    for (int s = t; s < S; s += 256) m = fmaxf(m, sc[s]);
    red[t] = m; __syncthreads();
    for (int off = 128; off; off >>= 1) {
        if (t < off) red[t] = fmaxf(red[t], red[t + off]);
        __syncthreads();
    }
    const float mx = red[0]; __syncthreads();

    float sum = 0.f;
    for (int s = t; s < S; s += 256) sum += __expf(sc[s] - mx);
    red[t] = sum; __syncthreads();
    for (int off = 128; off; off >>= 1) {
        if (t < off) red[t] += red[t + off];
        __syncthreads();
    }
    const float inv = 1.0f / red[0];

    for (int s = t; s < S; s += 256)
        c[(size_t)b * S + s] = __expf(sc[s] - mx) * inv * r[(size_t)b * S + s];
}

// ---------------------------------------------------------------------------
// Kernel 4: streaming pass 2 over x. grid (SPLIT, B), block 256 (float4 => D).
// partial[(b*SPLIT+sp), d] = sum over this split's s of c[b,s]*x[b,s,d]
// ---------------------------------------------------------------------------
__global__ void k_pass2(const float* __restrict__ x, const float* __restrict__ c,
                        float* __restrict__ partial) {
    const int sp = blockIdx.x, b = blockIdx.y, t = threadIdx.x;
    const int chunk = S / SPLIT;                         // 64 rows per block
    const float* xb = x + ((size_t)b * S + (size_t)sp * chunk) * D;
    const float* cb = c + (size_t)b * S + (size_t)sp * chunk;
    float4 acc = make_float4(0.f, 0.f, 0.f, 0.f);
    for (int s = 0; s < chunk; ++s) {
        if (s + 8 < chunk)   // CDNA5 global_prefetch_b8, 8 rows (32KB) ahead
            __builtin_prefetch(xb + (size_t)(s + 8) * D + t * 4, 0, 1);
        const float cv = cb[s];
        float4 v = ((const float4*)(xb + (size_t)s * D))[t];
        acc.x += cv * v.x; acc.y += cv * v.y; acc.z += cv * v.z; acc.w += cv * v.w;
    }
    ((float4*)(partial + ((size_t)(b * SPLIT + sp)) * D))[t] = acc;
}

// ---------------------------------------------------------------------------
// Kernel 5: reduce partials -> pooled[b,d] (with ln_w applied). grid B, block 256.
// ---------------------------------------------------------------------------
__global__ void k_reduce(const float* __restrict__ partial, const float* __restrict__ ln_w,
                         float* __restrict__ pooled) {
    const int b = blockIdx.x, t = threadIdx.x;
    float4 acc = make_float4(0.f, 0.f, 0.f, 0.f);
    for (int sp = 0; sp < SPLIT; ++sp) {
        float4 v = ((const float4*)(partial + ((size_t)(b * SPLIT + sp)) * D))[t];
        acc.x += v.x; acc.y += v.y; acc.z += v.z; acc.w += v.w;
    }
    float4 w = ((const float4*)ln_w)[t];
    acc.x *= w.x; acc.y *= w.y; acc.z *= w.z; acc.w *= w.w;
    ((float4*)(pooled + (size_t)b * D))[t] = acc;
}

// ---------------------------------------------------------------------------
// Kernel 6: GeGLU readout via V_WMMA_F32_16X16X4_F32 (full fp32 precision).
// out = gelu(pooled@W1 + b1) * (pooled@W2 + b2), M=32 N=1024 K=1024.
// 8 waves/block, 1 (mtile,ntile) per wave; 2*64 = 128 tiles -> grid 16.
// ---------------------------------------------------------------------------
__device__ __forceinline__ float gelu_tanh(float v) {
    return 0.5f * v * (1.0f + tanhf(0.7978845608028654f * (v + 0.044715f * v * v * v)));
}

__global__ void k_geglu(const float* __restrict__ pooled,
                        const float* __restrict__ W1, const float* __restrict__ b1,
                        const float* __restrict__ W2, const float* __restrict__ b2,
                        float* __restrict__ out) {
    const int wave = threadIdx.x >> 5;
    const int lane = threadIdx.x & 31;
    const int tile = blockIdx.x * 8 + wave;     // 0..127
    const int mt = tile >> 6;                   // 0..1
    const int nt = tile & 63;                   // 0..63
    const int m0 = mt * 16, n0 = nt * 16;
    const int half = lane >> 4;                 // A: K pair select; B: K pair select
    const int idx  = lane & 15;                 // A: M row; B: N col

    v8f acc1 = {}; v8f acc2 = {};
    for (int k = 0; k < D; k += 4) {
        const int kk = k + half * 2;
        v2f a, wv1, wv2;
        // A (16x4): lane holds pooled[m0+idx][kk], [kk+1]
        a.x = pooled[(size_t)(m0 + idx) * D + kk];
        a.y = pooled[(size_t)(m0 + idx) * D + kk + 1];
        // B (4x16): lane holds W[kk][n0+idx], W[kk+1][n0+idx]
        wv1.x = W1[(size_t)kk * DOUT + n0 + idx];
        wv1.y = W1[(size_t)(kk + 1) * DOUT + n0 + idx];
        wv2.x = W2[(size_t)kk * DOUT + n0 + idx];
        wv2.y = W2[(size_t)(kk + 1) * DOUT + n0 + idx];
        acc1 = __builtin_amdgcn_wmma_f32_16x16x4_f32(false, a, false, wv1,
                                                     (short)0, acc1, false, false);
        acc2 = __builtin_amdgcn_wmma_f32_16x16x4_f32(false, a, false, wv2,
                                                     (short)0, acc2, false, false);
    }
    // C/D layout: vgpr v -> M = m0 + v + 8*half, N = n0 + idx
    #pragma unroll
    for (int v = 0; v < 8; ++v) {
        const int m = m0 + v + half * 8;
        const int n = n0 + idx;
        const float h1 = acc1[v] + b1[n];
        const float h2 = acc2[v] + b2[n];
        out[(size_t)m * DOUT + n] = gelu_tanh(h1) * h2;
    }
}

// ---------------------------------------------------------------------------
extern "C" void kernel_launch(void* const* d_in, const int* in_sizes, int n_in,
                              void* d_out, int out_size, void* d_ws, size_t ws_size,
                              hipStream_t stream) {
    const float* x    = (const float*)d_in[0];   // [B,S,D]
    const float* ln_w = (const float*)d_in[1];   // [D]
    const float* Wk   = (const float*)d_in[2];   // [D,DIN]
    const float* q    = (const float*)d_in[3];   // [DIN]
    const float* W1   = (const float*)d_in[4];   // [D,DOUT]
    const float* b1   = (const float*)d_in[5];   // [DOUT]
    const float* W2   = (const float*)d_in[6];   // [D,DOUT]
    const float* b2   = (const float*)d_in[7];   // [DOUT]
    float* out = (float*)d_out;                  // [B,DOUT]

    // Workspace layout (floats), all 16B aligned:
    float* ws      = (float*)d_ws;
    float* qkw     = ws;                                   // 1024
    float* r       = qkw + D;                              // B*S
    float* score   = r + (size_t)B * S;                    // B*S
    float* c       = score + (size_t)B * S;                // B*S
    float* partial = c + (size_t)B * S;                    // SPLIT*B*D (4 MB)
    float* pooled  = partial + (size_t)SPLIT * B * D;      // B*D

    k_qkw    <<<D / 8, 256, 0, stream>>>(Wk, q, ln_w, qkw);
    k_pass1  <<<(B * S) / 8, 256, 0, stream>>>(x, qkw, r, score);
    k_softmax<<<B, 256, 0, stream>>>(score, r, c);
    k_pass2  <<<dim3(SPLIT, B), 256, 0, stream>>>(x, c, partial);
    k_reduce <<<B, 256, 0, stream>>>(partial, ln_w, pooled);
    k_geglu  <<<16, 256, 0, stream>>>(pooled, W1, b1, W2, b2, out);
}